// QuadraticWeightedKappa_39049842655437
// MI455X (gfx1250) — compile-verified
//
#include <hip/hip_runtime.h>

// Quadratic Weighted Kappa, N = 4,194,304 rows x 5 classes.
// Phase 1 (bandwidth-bound ~4us @ 23.3 TB/s): stream logits with b128 loads,
//   per-thread register counters (num, ct[5], cp[5]), wave32 shuffle reduce,
//   one 11-uint partial per block -> d_ws.
// Phase 2 (single wave32): reduce partials, compute denominator via
//   v_wmma_f32_16x16x4_f32 outer product of marginals, write -kappa.

typedef __attribute__((ext_vector_type(2))) float v2f;
typedef __attribute__((ext_vector_type(8))) float v8f;

#define NPART 11        // ct[5], cp[5], num
#define HIST_BLOCKS 1024
#define HIST_THREADS 256

__device__ __forceinline__ int argmax5(float a0, float a1, float a2, float a3, float a4) {
    int idx = 0; float best = a0;
    if (a1 > best) { best = a1; idx = 1; }
    if (a2 > best) { best = a2; idx = 2; }
    if (a3 > best) { best = a3; idx = 3; }
    if (a4 > best) { best = a4; idx = 4; }
    return idx;  // first max index, matches jnp.argmax
}

__global__ void qwk_hist(const float* __restrict__ logits,
                         const int* __restrict__ targets,
                         unsigned* __restrict__ partials, int n) {
    unsigned ct[5] = {0,0,0,0,0};
    unsigned cp[5] = {0,0,0,0,0};
    unsigned num = 0;

    const int tid    = blockIdx.x * blockDim.x + threadIdx.x;
    const int stride = gridDim.x * blockDim.x;
    const int G      = n >> 2;                      // groups of 4 rows (20 floats, 16B aligned)
    const float4* __restrict__ in4 = (const float4*)logits;
    const int4*   __restrict__ tg4 = (const int4*)targets;

    for (int g = tid; g < G; g += stride) {
        // Prefetch a few strides ahead (speculative; OOB prefetch is dropped).
        __builtin_prefetch(&in4[5ll * (g + 2 * stride)], 0, 0);

        float4 La = in4[5 * g + 0];
        float4 Lb = in4[5 * g + 1];
        float4 Lc = in4[5 * g + 2];
        float4 Ld = in4[5 * g + 3];
        float4 Le = in4[5 * g + 4];
        int4 t4 = tg4[g];

        int p0 = argmax5(La.x, La.y, La.z, La.w, Lb.x);
        int p1 = argmax5(Lb.y, Lb.z, Lb.w, Lc.x, Lc.y);
        int p2 = argmax5(Lc.z, Lc.w, Ld.x, Ld.y, Ld.z);
        int p3 = argmax5(Ld.w, Le.x, Le.y, Le.z, Le.w);
        int t0 = t4.x, t1 = t4.y, t2 = t4.z, t3 = t4.w;

        #pragma unroll
        for (int k = 0; k < 5; ++k) {
            ct[k] += (unsigned)((t0 == k) + (t1 == k) + (t2 == k) + (t3 == k));
            cp[k] += (unsigned)((p0 == k) + (p1 == k) + (p2 == k) + (p3 == k));
        }
        int d0 = t0 - p0, d1 = t1 - p1, d2 = t2 - p2, d3 = t3 - p3;
        num += (unsigned)(d0 * d0 + d1 * d1 + d2 * d2 + d3 * d3);
    }

    // Tail rows (n not divisible by 4) handled by the first few global threads.
    int rem = n & 3;
    if (tid < rem) {
        int r = (G << 2) + tid;
        float v0 = logits[5 * r + 0], v1 = logits[5 * r + 1], v2 = logits[5 * r + 2];
        float v3 = logits[5 * r + 3], v4 = logits[5 * r + 4];
        int p = argmax5(v0, v1, v2, v3, v4);
        int t = targets[r];
        #pragma unroll
        for (int k = 0; k < 5; ++k) { ct[k] += (unsigned)(t == k); cp[k] += (unsigned)(p == k); }
        int d = t - p; num += (unsigned)(d * d);
    }

    // wave32 reduction of the 11 counters
    #pragma unroll
    for (int off = 16; off > 0; off >>= 1) {
        #pragma unroll
        for (int k = 0; k < 5; ++k) {
            ct[k] += __shfl_down(ct[k], off);
            cp[k] += __shfl_down(cp[k], off);
        }
        num += __shfl_down(num, off);
    }

    __shared__ unsigned s[NPART];
    if (threadIdx.x < NPART) s[threadIdx.x] = 0u;
    __syncthreads();
    if ((threadIdx.x & 31) == 0) {       // one lane per wave
        #pragma unroll
        for (int k = 0; k < 5; ++k) {
            atomicAdd(&s[k], ct[k]);
            atomicAdd(&s[5 + k], cp[k]);
        }
        atomicAdd(&s[10], num);
    }
    __syncthreads();
    if (threadIdx.x < NPART)
        partials[blockIdx.x * NPART + threadIdx.x] = s[threadIdx.x];
}

// One wave32. EXEC is all-ones at the WMMA as required.
__global__ void qwk_final(const unsigned* __restrict__ partials,
                          float* __restrict__ out, int nblocks, int n) {
    __shared__ float sums[NPART];
    __shared__ float red[32];
    const int lane = threadIdx.x;       // 0..31

    if (lane < NPART) {
        unsigned long long acc = 0;
        for (int b = 0; b < nblocks; ++b) acc += partials[b * NPART + lane];
        sums[lane] = (float)acc;
    }
    __syncthreads();

    // A (16x4 f32): every K-column = mt  -> lanes 0-15 hold M=0..15 (K=0,1),
    // lanes 16-31 hold same M rows (K=2,3). So set both VGPRs = mt[lane&15].
    v2f a;
    float mt = ((lane & 15) < 5) ? sums[lane & 15] : 0.0f;
    a.x = mt; a.y = mt;
    // B (4x16 f32): exactly one K-row nonzero = mp (row striped across lanes
    // 0-15 in VGPR0). D[i][j] = sum_k A[i][k]*B[k][j] = mt[i]*mp[j] no matter
    // which K that row maps to, since all A columns are identical.
    v2f b;
    b.x = (lane < 5) ? sums[5 + lane] : 0.0f;
    b.y = 0.0f;
    v8f c = {};
    v8f d = __builtin_amdgcn_wmma_f32_16x16x4_f32(
        /*neg_a=*/false, a, /*neg_b=*/false, b,
        /*c_mod=*/(short)0, c, /*reuse_a=*/false, /*reuse_b=*/false);

    // C/D layout: lanes 0-15: N=lane, M=vgpr k; lanes 16-31: N=lane-16, M=k+8.
    const int Nn = lane & 15;
    const int Mbase = (lane < 16) ? 0 : 8;
    float partial = 0.0f;
    #pragma unroll
    for (int k = 0; k < 8; ++k) {
        float diff = (float)(Mbase + k - Nn);
        partial += d[k] * diff * diff;       // D * (i-j)^2
    }
    red[lane] = partial;
    __syncthreads();

    if (lane == 0) {
        float denom_scaled = 0.0f;
        for (int i = 0; i < 32; ++i) denom_scaled += red[i];
        float nf = (float)n;
        float numerator   = sums[10] / (16.0f * nf);            // sum (t-p)^2 / ((c-1)^2 n)
        float denominator = denom_scaled / (16.0f * nf * nf);   // mt^T W mp, normalized
        float kappa = 1.0f - numerator / (denominator + 1e-7f);
        out[0] = -kappa;
    }
}

extern "C" void kernel_launch(void* const* d_in, const int* in_sizes, int n_in,
                              void* d_out, int out_size, void* d_ws, size_t ws_size,
                              hipStream_t stream) {
    const float* logits  = (const float*)d_in[0];
    const int*   targets = (const int*)d_in[1];
    const int n = in_sizes[1];                 // N rows
    unsigned* partials = (unsigned*)d_ws;      // HIST_BLOCKS * NPART uints (~45 KB)

    qwk_hist<<<HIST_BLOCKS, HIST_THREADS, 0, stream>>>(logits, targets, partials, n);
    qwk_final<<<1, 32, 0, stream>>>(partials, (float*)d_out, HIST_BLOCKS, n);
}